// subclass_loss_33483565040216
// MI455X (gfx1250) — compile-verified
//
#include <hip/hip_runtime.h>
#include <hip/hip_bf16.h>

// Problem constants (fixed by reference)
#define BATCH   128
#define C_IN    512
#define OUTC    256     // == LDA_COMP
#define PX      64      // H*W
#define NROWS   8192    // BATCH*PX
#define EMBD    4096
#define SUB     32      // each_subclass = 4096/128
#define DD      4096
#define CH      512     // d-chunk for KL staging
#define NCH     8       // DD/CH

typedef __attribute__((ext_vector_type(2))) float v2f;
typedef __attribute__((ext_vector_type(8))) float v8f;

// f32 WMMA: D(16x16) = A(16x4) * B(4x16) + C. 8-arg VOP3P form.
__device__ __forceinline__ v8f wmma_f32x4(v2f a, v2f b, v8f c) {
  return __builtin_amdgcn_wmma_f32_16x16x4_f32(
      /*neg_a=*/false, a, /*neg_b=*/false, b,
      /*c_mod=*/(short)0, c, /*reuse_a=*/false, /*reuse_b=*/false);
}

// ---------------------------------------------------------------------------
// Kernel 1 (fused): per image b:
//   ft_b[64][256] = X_b[64][512] @ W^T + bias   (WMMA f32, staged to LDS)
//   G[64][32]     = ft_b @ Csel^T               (WMMA f32, Csel in LDS)
//   enc[b*64+p]   = label*32 + argmin_j (||c_j||^2 - 2*G[p][j])
// ---------------------------------------------------------------------------
__global__ __launch_bounds__(256)
void fused_lda_dist(const float* __restrict__ X,       // [128][512][64]
                    const float* __restrict__ W,       // [256][512]
                    const float* __restrict__ bias,    // [256]
                    const float* __restrict__ centers, // [4096][256]
                    const int*   __restrict__ labels,  // [128]
                    int*         __restrict__ enc)     // [8192]
{
  __shared__ float fta[PX][OUTC + 1];   // 64*257*4 = 65792 B
  __shared__ float cs[SUB][OUTC + 1];   // 32*257*4 = 32896 B
  __shared__ float cn2[SUB];
  __shared__ float candV[2][PX];
  __shared__ int   candI[2][PX];

  const int b    = blockIdx.x;
  const int tid  = threadIdx.x;
  const int wave = tid >> 5;
  const int lane = tid & 31;
  const int m    = lane & 15;      // column-of-tile / row-of-A lane index
  const int lh   = lane >> 4;      // lane half selects K pair / M+8

  // stage the 32 in-class centers
  const int base = labels[b] * SUB;
  for (int i = tid; i < SUB * OUTC; i += 256) {
    const int r = i >> 8, c = i & 255;
    cs[r][c] = centers[(size_t)(base + r) * OUTC + c];
  }
  __syncthreads();
  if (tid < SUB) {                 // ||c_j||^2 (lane-stride-257 -> conflict free)
    float s = 0.f;
    for (int c = 0; c < OUTC; ++c) { const float v = cs[tid][c]; s += v * v; }
    cn2[tid] = s;
  }

  // ---- GEMM1: wave w -> M-tile (w&3), col block ((w>>2)*128), 8 N-tiles
  const int mt = wave & 3;
  const int ob = (wave >> 2) * 128;
  const float* Xb = X + (size_t)b * C_IN * PX;
  const int hw = mt * 16 + m;      // pixel row handled by this lane

  v8f acc[8] = {};
  for (int k0 = 0; k0 < C_IN; k0 += 4) {
    const int ka = k0 + 2 * lh;
    v2f a;
    a.x = Xb[(size_t)ka * PX + hw];
    a.y = Xb[(size_t)(ka + 1) * PX + hw];
#pragma unroll
    for (int t = 0; t < 8; ++t) {
      const float* wp = W + (size_t)(ob + t * 16 + m) * C_IN + ka;
      v2f bb; bb.x = wp[0]; bb.y = wp[1];
      acc[t] = wmma_f32x4(a, bb, acc[t]);
    }
  }
  // C/D layout: acc[t][r] holds (M = r + 8*lh, N = m) of tile t
#pragma unroll
  for (int t = 0; t < 8; ++t) {
#pragma unroll
    for (int r = 0; r < 8; ++r) {
      const int col = ob + t * 16 + m;
      fta[mt * 16 + r + 8 * lh][col] = acc[t][r] + bias[col];
    }
  }
  __syncthreads();

  // ---- GEMM2: 4 M-tiles x 2 N-tiles, one 16x16 tile per wave, K = 256
  const int mt2 = wave & 3;
  const int jt  = wave >> 2;       // 0: cols 0-15, 1: cols 16-31
  v8f g = {};
  for (int k0 = 0; k0 < OUTC; k0 += 4) {
    const int ka = k0 + 2 * lh;
    v2f a;  a.x  = fta[mt2 * 16 + m][ka]; a.y  = fta[mt2 * 16 + m][ka + 1];
    v2f bb; bb.x = cs[jt * 16 + m][ka];   bb.y = cs[jt * 16 + m][ka + 1];
    g = wmma_f32x4(a, bb, g);
  }

  // argmin over the 16 cols of this tile (value = cn2[j] - 2*G; ties -> low j)
#pragma unroll
  for (int r = 0; r < 8; ++r) {
    const int Mrow = mt2 * 16 + r + 8 * lh;
    float v  = cn2[jt * 16 + m] - 2.f * g[r];
    int   ix = jt * 16 + m;
#pragma unroll
    for (int off = 8; off >= 1; off >>= 1) {   // butterfly within 16-lane half
      const float ov = __shfl_xor(v, off, 32);
      const int   oi = __shfl_xor(ix, off, 32);
      if (ov < v || (ov == v && oi < ix)) { v = ov; ix = oi; }
    }
    if (m == 0) { candV[jt][Mrow] = v; candI[jt][Mrow] = ix; }
  }
  __syncthreads();
  if (tid < PX) {
    const float v0 = candV[0][tid]; const int i0 = candI[0][tid];
    const float v1 = candV[1][tid]; const int i1 = candI[1][tid];
    enc[b * PX + tid] = base + ((v1 < v0) ? i1 : i0);  // tie -> lower index
  }
}

// ---------------------------------------------------------------------------
// Kernel 2: KL partials. Block = (chunk, b). scores[b] chunk [512][64] is
// streamed contiguously into a padded-transposed LDS tile; each wave owns 8
// pixels so teacher-row gathers are contiguous 128B per wave.
// part[row][chunk] = {max, sumexp, sum_t, sum_t_s, sum_xlogy}
// ---------------------------------------------------------------------------
__global__ __launch_bounds__(256)
void kl_partials(const float* __restrict__ scores,   // [128][4096][64]
                 const float* __restrict__ teacher,  // [4096][4096]
                 const int*   __restrict__ enc,      // [8192]
                 float*       __restrict__ part)     // [8192][8][5]
{
  __shared__ float tile[PX][CH + 1];   // 64*513*4 = 131328 B
  __shared__ int   encs[PX];

  const int chunk = blockIdx.x;        // 0..7
  const int b     = blockIdx.y;        // 0..127
  const int tid   = threadIdx.x;
  const int wave  = tid >> 5;
  const int lane  = tid & 31;

  if (tid < PX) encs[tid] = enc[b * PX + tid];

  const float* sb = scores + (size_t)b * DD * PX + (size_t)chunk * CH * PX;
  for (int i = tid; i < CH * PX; i += 256) {   // fully coalesced stream
    const int dl = i >> 6, p = i & 63;
    tile[p][dl] = sb[i];                       // stride-513 pad: no bank conflict
  }
  __syncthreads();

#pragma unroll
  for (int q = 0; q < 8; ++q) {
    const int p = wave * 8 + q;
    const float* trow = teacher + (size_t)encs[p] * DD + (size_t)chunk * CH;
    float mx = -3.0e38f, se = 0.f, St = 0.f, Sts = 0.f, Stlt = 0.f;
    for (int kk = lane; kk < CH; kk += 32) {
      const float s = tile[p][kk];
      const float t = trow[kk];                // contiguous across the wave
      const float mn = fmaxf(mx, s);           // online max for sumexp
      se = se * __expf(mx - mn) + __expf(s - mn);
      mx = mn;
      St += t;
      Sts  = fmaf(t, s, Sts);
      Stlt += (t > 0.f) ? t * __logf(t) : 0.f; // xlogy semantics
    }
    // reduce across the 32 lanes of this wave
    for (int off = 16; off >= 1; off >>= 1) {
      const float om  = __shfl_xor(mx, off, 32);
      const float ose = __shfl_xor(se, off, 32);
      const float M   = fmaxf(mx, om);
      se = se * __expf(mx - M) + ose * __expf(om - M);
      mx = M;
      St   += __shfl_xor(St,   off, 32);
      Sts  += __shfl_xor(Sts,  off, 32);
      Stlt += __shfl_xor(Stlt, off, 32);
    }
    if (lane == 0) {
      float* pp = part + (((size_t)(b * PX + p)) * NCH + chunk) * 5;
      pp[0] = mx; pp[1] = se; pp[2] = St; pp[3] = Sts; pp[4] = Stlt;
    }
  }
}

// ---------------------------------------------------------------------------
// Kernel 3: combine the 8 chunk partials per row (fixed order -> deterministic)
// ---------------------------------------------------------------------------
__global__ __launch_bounds__(256)
void kl_combine(const float* __restrict__ part, float* __restrict__ rowloss)
{
  const int n = blockIdx.x * blockDim.x + threadIdx.x;
  if (n >= NROWS) return;
  const float* pp = part + (size_t)n * NCH * 5;
  float mx = pp[0], se = pp[1], St = pp[2], Sts = pp[3], Stlt = pp[4];
  for (int c = 1; c < NCH; ++c) {
    const float* q = pp + c * 5;
    const float M = fmaxf(mx, q[0]);
    se = se * __expf(mx - M) + q[1] * __expf(q[0] - M);
    mx = M;
    St += q[2]; Sts += q[3]; Stlt += q[4];
  }
  const float lse = mx + __logf(se);
  rowloss[n] = Stlt - Sts + St * lse;
}

// ---------------------------------------------------------------------------
// Kernel 4: deterministic tree reduction -> scalar loss (batchmean over N)
// ---------------------------------------------------------------------------
__global__ __launch_bounds__(256)
void final_reduce(const float* __restrict__ rowloss, float* __restrict__ out)
{
  __shared__ float s[256];
  float a = 0.f;
  for (int i = threadIdx.x; i < NROWS; i += 256) a += rowloss[i];
  s[threadIdx.x] = a;
  __syncthreads();
  for (int st = 128; st >= 1; st >>= 1) {
    if (threadIdx.x < st) s[threadIdx.x] += s[threadIdx.x + st];
    __syncthreads();
  }
  if (threadIdx.x == 0) out[0] = s[0] / (float)NROWS;
}

// ---------------------------------------------------------------------------
extern "C" void kernel_launch(void* const* d_in, const int* in_sizes, int n_in,
                              void* d_out, int out_size, void* d_ws, size_t ws_size,
                              hipStream_t stream) {
  const float* feature  = (const float*)d_in[0];  // [128,512,8,8]
  const float* scores   = (const float*)d_in[1];  // [128,4096,8,8]
  const int*   labels   = (const int*)  d_in[2];  // [128]
  const float* ldaw     = (const float*)d_in[3];  // [256,512]
  const float* ldab     = (const float*)d_in[4];  // [256]
  const float* centers  = (const float*)d_in[5];  // [4096,256]
  const float* teacher  = (const float*)d_in[6];  // [4096,4096]
  float* out = (float*)d_out;

  // workspace layout: enc (32KB) | part (1.25MB) | rowloss (32KB)
  char* ws = (char*)d_ws;
  int*   enc     = (int*)  ws;
  float* part    = (float*)(ws + 32768);
  float* rowloss = (float*)(ws + 32768 + (size_t)NROWS * NCH * 5 * sizeof(float));

  fused_lda_dist<<<BATCH, 256, 0, stream>>>(feature, ldaw, ldab, centers, labels, enc);
  kl_partials<<<dim3(NCH, BATCH), 256, 0, stream>>>(scores, teacher, enc, part);
  kl_combine<<<NROWS / 256, 256, 0, stream>>>(part, rowloss);
  final_reduce<<<1, 256, 0, stream>>>(rowloss, out);
}